// chamferDist_89404039233870
// MI455X (gfx1250) — compile-verified
//
#include <hip/hip_runtime.h>

typedef __attribute__((ext_vector_type(2))) float v2f;
typedef __attribute__((ext_vector_type(8))) float v8f;

#define WAVES_PER_BLOCK 8

// Pack each 16-point tile directly in WMMA B-operand register layout:
// per tile, 32 float2 entries; entry[lane] is what lane needs for the
// 4x16 f32 B matrix (lanes 0-15 hold K=0,1 of col=lane; lanes 16-31 hold
// K=2,3 of col=lane-16):
//   lane <16 : (-2x, -2y)   of point tile*16 + lane
//   lane>=16 : (-2z,  s)    of point tile*16 + (lane-16),  s = x^2+y^2+z^2
// One global_load_b64 per lane per tile -> zero per-iteration VALU fixup.
__global__ void chamfer_prep_kernel(const float* __restrict__ xyz,
                                    float2* __restrict__ Bfmt,
                                    int npts, int total) {
    int i = blockIdx.x * blockDim.x + threadIdx.x;
    if (i >= total) return;
    int batch = i / npts;
    int j     = i % npts;
    int tile  = j >> 4;
    int pos   = j & 15;
    float x = xyz[3 * i + 0];
    float y = xyz[3 * i + 1];
    float z = xyz[3 * i + 2];
    size_t base = (size_t)batch * 2 * npts + (size_t)tile * 32;
    Bfmt[base + pos]      = make_float2(-2.0f * x, -2.0f * y);
    Bfmt[base + 16 + pos] = make_float2(-2.0f * z, x * x + y * y + z * z);
}

// One wave owns one 16-row tile of the query set, loops over all 16-col
// tiles of the target set. Per tile pair, one V_WMMA_F32_16X16X4_F32:
//   A row n = [x, y, z, 1]
//   B col m = [-2x', -2y', -2z', s2]
//   C       = 0 (loop-invariant)
//   D[n][m] = s2[m] - 2 p1.p2
// Row-min accumulates in the 8 accumulator VGPRs; s1[n] is added once at
// the end (min commutes with a row-constant), then clamped at 0.
__global__ void __launch_bounds__(256) chamfer_min_kernel(
    const float2* __restrict__ Brow,   // packed array for the query set
    const float2* __restrict__ Bcol,   // packed array for the target set
    float* __restrict__ out, int N1, int N2) {
    const int lane   = threadIdx.x & 31;
    const int waveId = threadIdx.x >> 5;

    const int tilesPerBatch  = N1 >> 4;
    const int groupsPerBatch = tilesPerBatch / WAVES_PER_BLOCK;
    const int b    = blockIdx.x / groupsPerBatch;
    const int tile = (blockIdx.x % groupsPerBatch) * WAVES_PER_BLOCK + waveId;
    const int n0   = tile << 4;

    const int  r  = lane & 15;
    const bool hi = lane >= 16;

    // Build A (16x4 f32: lanes 0-15 -> K=0,1 of row=lane; lanes 16-31 ->
    // K=2,3 of row=lane-16) from the packed layout: undo the -2 scale,
    // K=3 column is 1.0.
    const float2* QrowBase = Brow + (size_t)b * 2 * N1 + (size_t)tile * 32;
    float2 pa = QrowBase[lane];
    v2f a;
    a.x = -0.5f * pa.x;                  // x | z
    a.y = hi ? 1.0f : (-0.5f * pa.y);    // y | 1

    const float2* Q = Bcol + (size_t)b * 2 * N2;

    float dmin[8];
#pragma unroll
    for (int i = 0; i < 8; ++i) dmin[i] = 3.402823466e38f;

    const v8f czero = {};

    const int mtiles = N2 >> 4;
    for (int mt = 0; mt < mtiles; ++mt) {
        float2 q = Q[(size_t)mt * 32 + lane];    // B operand, ready to use
        v2f bv;
        bv.x = q.x;
        bv.y = q.y;
        v8f d = __builtin_amdgcn_wmma_f32_16x16x4_f32(
            /*neg_a=*/false, a, /*neg_b=*/false, bv,
            /*c_mod=*/(short)0, czero, /*reuse_a=*/false, /*reuse_b=*/false);
#pragma unroll
        for (int i = 0; i < 8; ++i) dmin[i] = fminf(dmin[i], d[i]);
    }

    // Min across the 16 lanes of each half (rows per-VGPR, cols per-lane).
#pragma unroll
    for (int i = 0; i < 8; ++i) {
        float v = dmin[i];
        v = fminf(v, __shfl_xor(v, 1, 32));
        v = fminf(v, __shfl_xor(v, 2, 32));
        v = fminf(v, __shfl_xor(v, 4, 32));
        v = fminf(v, __shfl_xor(v, 8, 32));
        dmin[i] = v;
    }

    // Lane 0 writes rows 0..7, lane 16 writes rows 8..15.
    // s1[row i] lives at QrowBase[16 + i].y (the packed 's' field).
    if (r == 0) {
        const int rowOff = hi ? 8 : 0;
        float* o = out + (size_t)b * N1 + n0 + rowOff;
#pragma unroll
        for (int i = 0; i < 8; ++i) {
            float s1 = QrowBase[16 + rowOff + i].y;
            o[i] = fmaxf(dmin[i] + s1, 0.0f);
        }
    }
}

extern "C" void kernel_launch(void* const* d_in, const int* in_sizes, int n_in,
                              void* d_out, int out_size, void* d_ws, size_t ws_size,
                              hipStream_t stream) {
    const float* xyz1 = (const float*)d_in[0];
    const float* xyz2 = (const float*)d_in[1];

    const int B = 16, N = 4096, M = 4096;

    float2* P1 = (float2*)d_ws;              // 2*B*N float2 (1 MB)
    float2* P2 = P1 + (size_t)2 * B * N;     // 2*B*M float2 (1 MB)

    chamfer_prep_kernel<<<(B * N + 255) / 256, 256, 0, stream>>>(xyz1, P1, N, B * N);
    chamfer_prep_kernel<<<(B * M + 255) / 256, 256, 0, stream>>>(xyz2, P2, M, B * M);

    float* dist1 = (float*)d_out;            // [B, N]
    float* dist2 = dist1 + (size_t)B * N;    // [B, M]

    const int grid1 = B * (N / 16 / WAVES_PER_BLOCK);
    const int grid2 = B * (M / 16 / WAVES_PER_BLOCK);
    chamfer_min_kernel<<<grid1, 256, 0, stream>>>(P1, P2, dist1, N, M);
    chamfer_min_kernel<<<grid2, 256, 0, stream>>>(P2, P1, dist2, M, N);
}